// GCN_50611894616840
// MI455X (gfx1250) — compile-verified
//
#include <hip/hip_runtime.h>
#include <cstdint>

// ---------------- problem constants (from reference) ----------------
#define BGRAPH 8
#define NNODES 8192
#define NEDGES 262144
#define EPRIME (NEDGES + NNODES)   // self-loops appended
#define FEAT   128
#define NEG_SLOPE 0.2f

typedef __attribute__((ext_vector_type(2))) float v2f;
typedef __attribute__((ext_vector_type(8))) float v8f;

// ---------------- helpers ----------------
// Order-preserving float->int map (monotonic for non-NaN): lets us use
// GLOBAL_ATOMIC_MAX_I32 instead of a CAS loop for float max.
__device__ __forceinline__ int f32_ord(float f) {
  int i = __float_as_int(f);
  return i >= 0 ? i : (i ^ 0x7fffffff);
}
__device__ __forceinline__ float ord_f32(int i) {
  return __int_as_float(i >= 0 ? i : (i ^ 0x7fffffff));
}
// Guaranteed-native non-returning f32 atomic add (CDNA5 GLOBAL_ATOMIC_ADD_F32).
// Avoids any chance of a CAS-loop legalization on the hottest path.
__device__ __forceinline__ void atomic_add_f32(float* p, float v) {
  asm volatile("global_atomic_add_f32 %0, %1, off"
               :
               : "v"(p), "v"(v)
               : "memory");
}
__device__ __forceinline__ float leaky(float v) {
  return v >= 0.0f ? v : NEG_SLOPE * v;
}

// ---------------- WMMA f32 GEMM: H[g] = X[g] @ B (+bias) ----------------
// X: [B, NNODES, FEAT], W: [FEAT, FEAT] (TRANSB=false: B=W; TRANSB=true: B=W^T,
// i.e. H = X @ W^T for the final nn.Linear). Block = 256 thr = 8 waves; each
// wave computes one 16x16 tile via 32x v_wmma_f32_16x16x4_f32; block covers a
// 16-row x 128-col tile. Grid = (NNODES/16, B).
template <bool TRANSB, bool BIAS>
__global__ void __launch_bounds__(256)
gemm_wmma_f32(const float* __restrict__ X, const float* __restrict__ W,
              const float* __restrict__ bias, float* __restrict__ H) {
  const int g    = blockIdx.y;
  const int m0   = blockIdx.x * 16;
  const int wave = threadIdx.x >> 5;
  const int lane = threadIdx.x & 31;
  const int half = lane >> 4;          // 0: lanes 0-15, 1: lanes 16-31
  const int l    = lane & 15;
  const int n0   = wave * 16;

  const float* xrow = X + ((size_t)g * NNODES + (m0 + l)) * FEAT;

  v8f acc = {};
#pragma unroll
  for (int k = 0; k < FEAT; k += 4) {
    const int ka = k + 2 * half;       // A 16x4 f32 layout: K=0,1 | K=2,3 per half
    v2f a;
    a.x = xrow[ka];
    a.y = xrow[ka + 1];
    v2f b;                              // B 4x16 f32 layout mirrored across halves
    if (TRANSB) {
      const float* wr = W + (size_t)(n0 + l) * FEAT + ka; // B[k][n] = W[n][k]
      b.x = wr[0];
      b.y = wr[1];
    } else {
      b.x = W[(size_t)ka * FEAT + (n0 + l)];
      b.y = W[(size_t)(ka + 1) * FEAT + (n0 + l)];
    }
    acc = __builtin_amdgcn_wmma_f32_16x16x4_f32(
        /*neg_a=*/false, a, /*neg_b=*/false, b,
        /*c_mod=*/(short)0, acc, /*reuse_a=*/false, /*reuse_b=*/false);
  }

  const float bv = BIAS ? bias[n0 + l] : 0.0f;
  // D layout: VGPR i -> M = i + 8*half, N = n0 + l
  float* hout = H + ((size_t)g * NNODES + m0 + 8 * half) * FEAT + (n0 + l);
#pragma unroll
  for (int i = 0; i < 8; ++i)
    hout[(size_t)i * FEAT] = acc[i] + bv;
}

// ---------------- per-node attention logits: AS = h.a_src, AD = h.a_dst ----
// wave per node (flat over B*NNODES); 4 features per lane; wave32 shfl reduce.
__global__ void __launch_bounds__(256)
alpha_kernel(const float* __restrict__ H, const float* __restrict__ a_src,
             const float* __restrict__ a_dst, float* __restrict__ AS,
             float* __restrict__ AD) {
  const int wave = threadIdx.x >> 5;
  const int lane = threadIdx.x & 31;
  const int node = blockIdx.x * 8 + wave;      // flat [0, B*NNODES)
  const int f    = lane * 4;
  const float4 h4 = *(const float4*)(H + (size_t)node * FEAT + f);
  float ps = h4.x * a_src[f] + h4.y * a_src[f + 1] +
             h4.z * a_src[f + 2] + h4.w * a_src[f + 3];
  float pd = h4.x * a_dst[f] + h4.y * a_dst[f + 1] +
             h4.z * a_dst[f + 2] + h4.w * a_dst[f + 3];
#pragma unroll
  for (int off = 16; off > 0; off >>= 1) {
    ps += __shfl_xor(ps, off, 32);
    pd += __shfl_xor(pd, off, 32);
  }
  if (lane == 0) {
    AS[node] = ps;
    AD[node] = pd;
  }
}

// ---------------- init per-node segment state ----------------
__global__ void init_seg_kernel(int* __restrict__ MORD, float* __restrict__ SSUM) {
  const int i = blockIdx.x * blockDim.x + threadIdx.x;
  MORD[i] = (int)0x80000000;  // -inf in ordered-int space (self-loop always updates)
  SSUM[i] = 0.0f;
}

__device__ __forceinline__ void edge_nodes(const int* __restrict__ ei, int e,
                                           int& s, int& d) {
  if (e < NEDGES) {
    s = ei[e];
    d = ei[NEDGES + e];
  } else {
    s = d = e - NEDGES;   // appended self-loop
  }
}

// ---------------- edge pass 1: segment max (int-ordered atomic max) --------
__global__ void __launch_bounds__(256)
edge_max_kernel(const int* __restrict__ EI, const float* __restrict__ AS,
                const float* __restrict__ AD, int* __restrict__ MORD) {
  const long idx = (long)blockIdx.x * blockDim.x + threadIdx.x;
  const int g = (int)(idx / EPRIME);
  const int e = (int)(idx - (long)g * EPRIME);
  int s, d;
  edge_nodes(EI + (size_t)g * 2 * NEDGES, e, s, d);
  const float v = leaky(AS[g * NNODES + s] + AD[g * NNODES + d]);
  atomicMax(&MORD[g * NNODES + d], f32_ord(v));   // native global_atomic_max_i32
}

// ---------------- edge pass 2: segment sum of exp(e - m) -------------------
__global__ void __launch_bounds__(256)
edge_sum_kernel(const int* __restrict__ EI, const float* __restrict__ AS,
                const float* __restrict__ AD, const int* __restrict__ MORD,
                float* __restrict__ SSUM) {
  const long idx = (long)blockIdx.x * blockDim.x + threadIdx.x;
  const int g = (int)(idx / EPRIME);
  const int e = (int)(idx - (long)g * EPRIME);
  int s, d;
  edge_nodes(EI + (size_t)g * 2 * NEDGES, e, s, d);
  const int gd = g * NNODES + d;
  const float v = leaky(AS[g * NNODES + s] + AD[gd]);
  const float w = __expf(v - ord_f32(MORD[gd]));
  atomic_add_f32(&SSUM[gd], w);
}

// ---------------- edge pass 3: ACC[dst] += alpha * H[src] ------------------
// wave per edge; 4 features per lane; native f32 atomics into L2-resident ACC.
__global__ void __launch_bounds__(256)
edge_agg_kernel(const int* __restrict__ EI, const float* __restrict__ AS,
                const float* __restrict__ AD, const int* __restrict__ MORD,
                const float* __restrict__ SSUM, const float* __restrict__ H,
                float* __restrict__ ACC) {
  const int wave = threadIdx.x >> 5;
  const int lane = threadIdx.x & 31;
  const long widx = (long)blockIdx.x * 8 + wave;
  const int g = (int)(widx / EPRIME);
  const int e = (int)(widx - (long)g * EPRIME);
  int s, d;
  edge_nodes(EI + (size_t)g * 2 * NEDGES, e, s, d);
  const int gs = g * NNODES + s, gd = g * NNODES + d;
  const float v = leaky(AS[gs] + AD[gd]);
  const float alpha = __expf(v - ord_f32(MORD[gd])) / SSUM[gd];
  const int f = lane * 4;
  const float4 h4 = *(const float4*)(H + (size_t)gs * FEAT + f);
  float* o = ACC + (size_t)gd * FEAT + f;
  atomic_add_f32(o + 0, alpha * h4.x);
  atomic_add_f32(o + 1, alpha * h4.y);
  atomic_add_f32(o + 2, alpha * h4.z);
  atomic_add_f32(o + 3, alpha * h4.w);
}

// ---------------- bias + swish: X = swish(ACC + b) ----------------
__global__ void __launch_bounds__(256)
bias_swish_kernel(const float* __restrict__ ACC, const float* __restrict__ b,
                  float* __restrict__ X) {
  const long i = (long)blockIdx.x * blockDim.x + threadIdx.x;
  const int f = (int)(i & (FEAT - 1));
  const float v = ACC[i] + b[f];
  X[i] = v / (1.0f + __expf(-v));   // v * sigmoid(v)
}

// =====================================================================
extern "C" void kernel_launch(void* const* d_in, const int* in_sizes, int n_in,
                              void* d_out, int out_size, void* d_ws, size_t ws_size,
                              hipStream_t stream) {
  const float* x0   = (const float*)d_in[0];   // [B,N,128]
  const int*   ei   = (const int*)d_in[1];     // [B,2,E]
  const float* W1   = (const float*)d_in[2];
  const float* as1  = (const float*)d_in[3];
  const float* ad1  = (const float*)d_in[4];
  const float* b1   = (const float*)d_in[5];
  const float* W2   = (const float*)d_in[6];
  const float* as2  = (const float*)d_in[7];
  const float* ad2  = (const float*)d_in[8];
  const float* b2   = (const float*)d_in[9];
  const float* fcW  = (const float*)d_in[10];
  const float* fcb  = (const float*)d_in[11];
  float* out = (float*)d_out;                  // [B,N,128]

  const size_t BN      = (size_t)BGRAPH * NNODES;          // 65536
  const size_t BIGELTS = BN * FEAT;                        // 8.39M floats
  char* ws = (char*)d_ws;
  float* BUF0 = (float*)ws;                ws += BIGELTS * sizeof(float); // 33.5MB
  float* BUF1 = (float*)ws;                ws += BIGELTS * sizeof(float); // 33.5MB
  float* AS   = (float*)ws;                ws += BN * sizeof(float);
  float* AD   = (float*)ws;                ws += BN * sizeof(float);
  int*   MORD = (int*)ws;                  ws += BN * sizeof(int);
  float* SSUM = (float*)ws;                // total ~68 MB

  const dim3 gGemm(NNODES / 16, BGRAPH);                       // 512 x 8
  const int  gAlpha = (int)(BN / 8);                           // 8192
  const int  gInit  = (int)(BN / 256);                         // 256
  const int  gEdgeT = (int)(((long)BGRAPH * EPRIME) / 256);    // 8448 (exact)
  const int  gEdgeW = (int)(((long)BGRAPH * EPRIME) / 8);      // 270336 (exact)
  const int  gElem  = (int)(BIGELTS / 256);                    // 32768

  // ---------------- Layer 1: H=BUF0, ACC=BUF1, X1 -> BUF0 ----------------
  gemm_wmma_f32<false, false><<<gGemm, 256, 0, stream>>>(x0, W1, nullptr, BUF0);
  alpha_kernel<<<gAlpha, 256, 0, stream>>>(BUF0, as1, ad1, AS, AD);
  init_seg_kernel<<<gInit, 256, 0, stream>>>(MORD, SSUM);
  hipMemsetAsync(BUF1, 0, BIGELTS * sizeof(float), stream);
  edge_max_kernel<<<gEdgeT, 256, 0, stream>>>(ei, AS, AD, MORD);
  edge_sum_kernel<<<gEdgeT, 256, 0, stream>>>(ei, AS, AD, MORD, SSUM);
  edge_agg_kernel<<<gEdgeW, 256, 0, stream>>>(ei, AS, AD, MORD, SSUM, BUF0, BUF1);
  bias_swish_kernel<<<gElem, 256, 0, stream>>>(BUF1, b1, BUF0);   // X1 in BUF0

  // ---------------- Layer 2: H=BUF1, ACC=BUF0, X2 -> BUF1 ----------------
  gemm_wmma_f32<false, false><<<gGemm, 256, 0, stream>>>(BUF0, W2, nullptr, BUF1);
  alpha_kernel<<<gAlpha, 256, 0, stream>>>(BUF1, as2, ad2, AS, AD);
  init_seg_kernel<<<gInit, 256, 0, stream>>>(MORD, SSUM);
  hipMemsetAsync(BUF0, 0, BIGELTS * sizeof(float), stream);
  edge_max_kernel<<<gEdgeT, 256, 0, stream>>>(ei, AS, AD, MORD);
  edge_sum_kernel<<<gEdgeT, 256, 0, stream>>>(ei, AS, AD, MORD, SSUM);
  edge_agg_kernel<<<gEdgeW, 256, 0, stream>>>(ei, AS, AD, MORD, SSUM, BUF1, BUF0);
  bias_swish_kernel<<<gElem, 256, 0, stream>>>(BUF0, b2, BUF1);   // X2 in BUF1

  // ---------------- FC: out = X2 @ fcW^T + fcb ----------------
  gemm_wmma_f32<true, true><<<gGemm, 256, 0, stream>>>(BUF1, fcW, fcb, out);
}